// GNN_74088185856510
// MI455X (gfx1250) — compile-verified
//
#include <hip/hip_runtime.h>
#include <hip/hip_bf16.h>
#include <math.h>

#define N_NODES  100000
#define N_EDGES  1600000
#define N_GRAPHS 2000
#define F        64   // F_IN == HID == 64

typedef float v2f __attribute__((ext_vector_type(2)));
typedef float v8f __attribute__((ext_vector_type(8)));

// ---------------------------------------------------------------- utilities
__global__ void GNN_zero_kernel(float* __restrict__ p, int n) {
    int i = blockIdx.x * blockDim.x + threadIdx.x;
    if (i < n) p[i] = 0.0f;
}

// out_deg[src[e]] += 1 ; in_deg[dst[e]] += 1
__global__ void GNN_degree_kernel(const int* __restrict__ src,
                                  const int* __restrict__ dst,
                                  float* __restrict__ odeg,
                                  float* __restrict__ ideg) {
    int e = blockIdx.x * blockDim.x + threadIdx.x;
    if (e < N_EDGES) {
        atomicAdd(&odeg[src[e]], 1.0f);
        atomicAdd(&ideg[dst[e]], 1.0f);
    }
}

// d[i] = rsqrt(max(d[i], 1))  (applied to odeg and ideg jointly: 2*N elements)
__global__ void GNN_rsqrt_kernel(float* __restrict__ d, int n) {
    int i = blockIdx.x * blockDim.x + threadIdx.x;
    if (i < n) d[i] = rsqrtf(fmaxf(d[i], 1.0f));
}

// ---------------------------------------------------------------- WMMA GEMM
// out[r, :] = (h[r, :] * scale[r]) @ W       h: n_rows x 64, W: 64 x 64
// One wave computes a 16-row x 64-col tile via V_WMMA_F32_16X16X4_F32,
// K-loop of 16 steps (K=4 each), 4 column tiles (4 x v8f accumulators).
#define WAVES_PER_BLOCK 4
__global__ __launch_bounds__(32 * WAVES_PER_BLOCK)
void GNN_gemm_wmma_kernel(const float* __restrict__ h,
                          const float* __restrict__ scale,
                          const float* __restrict__ W,
                          float* __restrict__ out,
                          int n_rows) {
    // +1 padding (stride 65) so A-fragment column reads (fixed k, varying m)
    // hit distinct LDS banks instead of a 16-way conflict.
    __shared__ float sW[64 * 65];
    __shared__ float sh[WAVES_PER_BLOCK][16 * 65];

    const int lane = threadIdx.x & 31;
    const int wave = threadIdx.x >> 5;

    // stage W (64x64) into LDS, padded
    for (int i = threadIdx.x; i < 64 * 64; i += 32 * WAVES_PER_BLOCK) {
        int r = i >> 6, c = i & 63;
        sW[r * 65 + c] = W[i];
    }

    const int rowTile = blockIdx.x * WAVES_PER_BLOCK + wave;
    const int row0    = rowTile * 16;
    const bool active = (row0 < n_rows);      // wave-uniform (n_rows % 16 == 0)

    if (active) {
        // stage 16x64 row tile, pre-scaled by out-degree norm
        for (int i = lane; i < 16 * 64; i += 32) {
            int r = i >> 6, c = i & 63;
            sh[wave][r * 65 + c] = h[(size_t)(row0 + r) * F + c] * scale[row0 + r];
        }
    }
    __syncthreads();

    if (active) {
        v8f c0 = {}, c1 = {}, c2 = {}, c3 = {};
        const int m  = lane & 15;            // A row / B col within tile
        const int n  = lane & 15;
        const int kb = (lane >> 4) * 2;      // lanes 0-15: K=0,1 ; lanes 16-31: K=2,3
        const float* shw = &sh[wave][0];

#pragma unroll
        for (int k0 = 0; k0 < 64; k0 += 4) {
            const int kA = k0 + kb;
            v2f a;
            a.x = shw[m * 65 + kA];
            a.y = shw[m * 65 + kA + 1];
            v2f b0, b1, b2, b3;
            b0.x = sW[kA * 65 + n +  0];  b0.y = sW[(kA + 1) * 65 + n +  0];
            b1.x = sW[kA * 65 + n + 16];  b1.y = sW[(kA + 1) * 65 + n + 16];
            b2.x = sW[kA * 65 + n + 32];  b2.y = sW[(kA + 1) * 65 + n + 32];
            b3.x = sW[kA * 65 + n + 48];  b3.y = sW[(kA + 1) * 65 + n + 48];
            c0 = __builtin_amdgcn_wmma_f32_16x16x4_f32(false, a, false, b0, (short)0, c0, false, false);
            c1 = __builtin_amdgcn_wmma_f32_16x16x4_f32(false, a, false, b1, (short)0, c1, false, false);
            c2 = __builtin_amdgcn_wmma_f32_16x16x4_f32(false, a, false, b2, (short)0, c2, false, false);
            c3 = __builtin_amdgcn_wmma_f32_16x16x4_f32(false, a, false, b3, (short)0, c3, false, false);
        }

        // D layout: VGPR r -> M = (lane<16 ? r : 8+r), N = lane&15
        const int rbase = (lane >> 4) * 8;
#pragma unroll
        for (int r = 0; r < 8; ++r) {
            size_t row = (size_t)(row0 + rbase + r) * F;
            out[row + n +  0] = c0[r];
            out[row + n + 16] = c1[r];
            out[row + n + 32] = c2[r];
            out[row + n + 48] = c3[r];
        }
    }
}

// ---------------------------------------------------------------- edge scatter
// m[dst[e]*64 + f] += hw[src[e]*64 + f]   (one thread per (edge, feature))
__global__ void GNN_scatter_kernel(const float* __restrict__ hw,
                                   const int* __restrict__ src,
                                   const int* __restrict__ dst,
                                   float* __restrict__ m) {
    long long t = (long long)blockIdx.x * blockDim.x + threadIdx.x;
    int e = (int)(t >> 6);
    int f = (int)(t & 63);
    if (e < N_EDGES) {
        int s = src[e], d = dst[e];
        atomicAdd(&m[(size_t)d * F + f], hw[(size_t)s * F + f]);
    }
}

// h[t] = (relu?) relu(m[t]*idi[node] + b[f])    (in-place safe)
__global__ void GNN_post_kernel(float* __restrict__ m,
                                const float* __restrict__ idi,
                                const float* __restrict__ b,
                                int apply_relu) {
    long long t = (long long)blockIdx.x * blockDim.x + threadIdx.x;
    int node = (int)(t >> 6);
    int f = (int)(t & 63);
    if (node < N_NODES) {
        float v = m[t] * idi[node] + b[f];
        m[t] = apply_relu ? fmaxf(v, 0.0f) : v;
    }
}

// ---------------------------------------------------------------- norm + pool
__global__ void GNN_norm_kernel(const float* __restrict__ h,
                                float* __restrict__ accum) {
    __shared__ float red[256];
    int node = blockIdx.x * 256 + threadIdx.x;
    float s = 0.0f;
    if (node < N_NODES) {
        float acc = 0.0f;
#pragma unroll
        for (int f = 0; f < F; ++f) {
            float v = h[(size_t)node * F + f];
            acc += v * v;
        }
        s = sqrtf(acc);
    }
    red[threadIdx.x] = s;
    __syncthreads();
    for (int off = 128; off > 0; off >>= 1) {
        if (threadIdx.x < off) red[threadIdx.x] += red[threadIdx.x + off];
        __syncthreads();
    }
    if (threadIdx.x == 0) atomicAdd(accum, red[0]);
}

// out[gid[node]*64+f] += h[node*64+f] * (sqrt(64)*N_NODES / sum_norm)
__global__ void GNN_pool_kernel(const float* __restrict__ h,
                                const int* __restrict__ gid,
                                const float* __restrict__ accum,
                                float* __restrict__ out) {
    long long t = (long long)blockIdx.x * blockDim.x + threadIdx.x;
    int node = (int)(t >> 6);
    int f = (int)(t & 63);
    if (node < N_NODES) {
        float factor = 8.0f * (float)N_NODES / accum[0];
        atomicAdd(&out[(size_t)gid[node] * F + f], h[(size_t)node * F + f] * factor);
    }
}

// ---------------------------------------------------------------- launcher
extern "C" void kernel_launch(void* const* d_in, const int* in_sizes, int n_in,
                              void* d_out, int out_size, void* d_ws, size_t ws_size,
                              hipStream_t stream) {
    const float* h_in   = (const float*)d_in[0];
    const int*   e_src  = (const int*)  d_in[1];
    const int*   e_dst  = (const int*)  d_in[2];
    const int*   g_ids  = (const int*)  d_in[3];
    const float* Ws[3]  = { (const float*)d_in[4], (const float*)d_in[6], (const float*)d_in[8] };
    const float* bs[3]  = { (const float*)d_in[5], (const float*)d_in[7], (const float*)d_in[9] };
    float* out = (float*)d_out;

    // workspace layout (bytes): bufA | bufB | odi | idi | norm_accum  (~52 MB)
    const size_t NF = (size_t)N_NODES * F;
    char* ws    = (char*)d_ws;
    float* bufA = (float*)(ws);
    float* bufB = (float*)(ws + NF * sizeof(float));
    float* odi  = (float*)(ws + 2 * NF * sizeof(float));
    float* idi  = odi + N_NODES;                       // contiguous with odi
    float* nacc = idi + N_NODES;

    const int TPB = 256;
    const int nfeat_blocks = (int)(((long long)N_NODES * F + TPB - 1) / TPB);
    const int efeat_blocks = (int)(((long long)N_EDGES * F + TPB - 1) / TPB);
    const int gemm_blocks  = (N_NODES / 16 + WAVES_PER_BLOCK - 1) / WAVES_PER_BLOCK;

    // degrees -> rsqrt norms
    GNN_zero_kernel<<<(2 * N_NODES + TPB - 1) / TPB, TPB, 0, stream>>>(odi, 2 * N_NODES);
    GNN_degree_kernel<<<(N_EDGES + TPB - 1) / TPB, TPB, 0, stream>>>(e_src, e_dst, odi, idi);
    GNN_rsqrt_kernel<<<(2 * N_NODES + TPB - 1) / TPB, TPB, 0, stream>>>(odi, 2 * N_NODES);

    // three GCN layers: gemm -> zero target -> scatter -> (idi,b,relu)
    const float* h_cur = h_in;
    for (int l = 0; l < 3; ++l) {
        GNN_gemm_wmma_kernel<<<gemm_blocks, 32 * WAVES_PER_BLOCK, 0, stream>>>(
            h_cur, odi, Ws[l], bufA, N_NODES);
        GNN_zero_kernel<<<nfeat_blocks, TPB, 0, stream>>>(bufB, (int)NF);
        GNN_scatter_kernel<<<efeat_blocks, TPB, 0, stream>>>(bufA, e_src, e_dst, bufB);
        GNN_post_kernel<<<nfeat_blocks, TPB, 0, stream>>>(bufB, idi, bs[l], (l < 2) ? 1 : 0);
        h_cur = bufB;
    }

    // factor = sqrt(64) / mean(||h||), then sum-pool per graph
    GNN_zero_kernel<<<1, TPB, 0, stream>>>(nacc, 1);
    GNN_norm_kernel<<<(N_NODES + 255) / 256, 256, 0, stream>>>(bufB, nacc);
    GNN_zero_kernel<<<(out_size + TPB - 1) / TPB, TPB, 0, stream>>>(out, out_size);
    GNN_pool_kernel<<<nfeat_blocks, TPB, 0, stream>>>(bufB, g_ids, nacc, out);
}